// GCN_26938034881120
// MI455X (gfx1250) — compile-verified
//
#include <hip/hip_runtime.h>

typedef float v2f __attribute__((ext_vector_type(2)));
typedef float v8f __attribute__((ext_vector_type(8)));

#define CH 128

// Guaranteed hardware f32 atomic add at device scope (no CAS fallback).
static __device__ __forceinline__ void atomicAddF32Dev(float* p, float v) {
  asm volatile("global_atomic_add_f32 %0, %1, off scope:SCOPE_DEV"
               :: "v"(p), "v"(v) : "memory");
}

__global__ __launch_bounds__(256) void k_init(int* __restrict__ degi,
                                              float* __restrict__ sums, int N) {
  int i = blockIdx.x * 256 + threadIdx.x;
  if (i < N) degi[i] = 1;          // self-loop contribution to degree
  if (i < 256) sums[i] = 0.0f;     // BN sum + sumsq
}

__global__ __launch_bounds__(256) void k_deg(const int* __restrict__ rows,
                                             int* __restrict__ degi, int E) {
  int e = blockIdx.x * 256 + threadIdx.x;
  if (e < E) atomicAdd(&degi[rows[e]], 1);
}

__global__ __launch_bounds__(256) void k_dis(const int* __restrict__ degi,
                                             float* __restrict__ dis, int N) {
  int n = blockIdx.x * 256 + threadIdx.x;
  if (n < N) dis[n] = rsqrtf((float)degi[n]);
}

// Out[n,j] = sum_k act(A[n,k]) * W[k,j].  One wave -> 16x128 output strip.
// A frag (f32 16x4): lanes 0-15 hold M=lane,K={k0,k0+1}; lanes 16-31 M=lane-16,K={k0+2,k0+3}.
// B frag (f32 4x16): VGPR0 = row k0 (low half) / k0+2 (high half); VGPR1 = k0+1 / k0+3.
template <bool APPLY_BN>
__global__ __launch_bounds__(256) void k_gemm(const float* __restrict__ A,
                                              const float* __restrict__ W,
                                              float* __restrict__ Out,
                                              const float* __restrict__ scale,
                                              const float* __restrict__ shift,
                                              int N) {
  const int strip = blockIdx.x * 8 + (threadIdx.x >> 5);
  const int nStrips = (N + 15) >> 4;
  if (strip >= nStrips) return;          // wave-uniform: EXEC all-1s for WMMA
  const int lane = threadIdx.x & 31;
  const int half = lane >> 4;
  const int lm   = lane & 15;
  const int row0 = strip * 16;
  int rA = row0 + lm;
  if (rA >= N) rA = N - 1;               // clamp ragged tail loads
  const float* arow = A + (size_t)rA * CH;

  v8f acc[8];
#pragma unroll
  for (int t = 0; t < 8; ++t)
#pragma unroll
    for (int j = 0; j < 8; ++j) acc[t][j] = 0.0f;

  const int kq = half * 2;
#pragma unroll 4
  for (int k0 = 0; k0 < CH; k0 += 4) {
    const int ka = k0 + kq;
    float ax = arow[ka];
    float ay = arow[ka + 1];
    if (APPLY_BN) {                      // fused batchnorm-affine + ReLU
      ax = fmaxf(fmaf(scale[ka], ax, shift[ka]), 0.0f);
      ay = fmaxf(fmaf(scale[ka + 1], ay, shift[ka + 1]), 0.0f);
    }
    v2f a; a.x = ax; a.y = ay;
#pragma unroll
    for (int t = 0; t < 8; ++t) {
      v2f b;
      b.x = W[(size_t)ka * CH + t * 16 + lm];
      b.y = W[(size_t)(ka + 1) * CH + t * 16 + lm];
      acc[t] = __builtin_amdgcn_wmma_f32_16x16x4_f32(
          false, a, false, b, (short)0, acc[t], false, false);
    }
  }
#pragma unroll
  for (int t = 0; t < 8; ++t) {
#pragma unroll
    for (int j = 0; j < 8; ++j) {
      const int r = row0 + half * 8 + j;
      if (r < N) Out[(size_t)r * CH + t * 16 + lm] = acc[t][j];
    }
  }
}

// out[n,:] = bias + dis[n]^2 * hw[n,:]   (self-loop term, plain stores)
__global__ __launch_bounds__(256) void k_seed(const float* __restrict__ hw,
                                              const float* __restrict__ dis,
                                              const float* __restrict__ bias,
                                              float* __restrict__ out, int N) {
  const int i = blockIdx.x * 256 + threadIdx.x;  // over N*32 float4 chunks
  if (i >= N * 32) return;
  const int n = i >> 5;
  const int q = (i & 31) * 4;
  const float d2 = dis[n] * dis[n];
  const float4 v = *(const float4*)(hw + (size_t)n * CH + q);
  const float4 b = *(const float4*)(bias + q);
  float4 o;
  o.x = fmaf(d2, v.x, b.x);
  o.y = fmaf(d2, v.y, b.y);
  o.z = fmaf(d2, v.z, b.z);
  o.w = fmaf(d2, v.w, b.w);
  *(float4*)(out + (size_t)n * CH + q) = o;
}

// out[row,:] += dis[row]*dis[col]*hw[col,:]; one wave per edge, float4 gather.
__global__ __launch_bounds__(256) void k_edges(const int* __restrict__ rows,
                                               const int* __restrict__ cols,
                                               const float* __restrict__ dis,
                                               const float* __restrict__ hw,
                                               float* __restrict__ out, int E) {
  const int e = (blockIdx.x * 256 + threadIdx.x) >> 5;
  if (e >= E) return;                    // wave-uniform
  const int lane = threadIdx.x & 31;
  const int r = rows[e];
  const int c = cols[e];
  const float nrm = dis[r] * dis[c];
  const float4 v = *(const float4*)(hw + (size_t)c * CH + lane * 4);
  float* o = out + (size_t)r * CH + lane * 4;
  atomicAddF32Dev(o + 0, nrm * v.x);
  atomicAddF32Dev(o + 1, nrm * v.y);
  atomicAddF32Dev(o + 2, nrm * v.z);
  atomicAddF32Dev(o + 3, nrm * v.w);
}

__global__ __launch_bounds__(256) void k_bnstats(const float* __restrict__ h,
                                                 float* __restrict__ sums, int N) {
  const int d  = threadIdx.x & 127;
  const int rg = threadIdx.x >> 7;       // 0..1
  const int stride = gridDim.x * 2;
  float s = 0.0f, ss = 0.0f;
  for (int n = blockIdx.x * 2 + rg; n < N; n += stride) {
    const float v = h[(size_t)n * CH + d];
    s += v;
    ss = fmaf(v, v, ss);
  }
  atomicAddF32Dev(&sums[d], s);
  atomicAddF32Dev(&sums[CH + d], ss);
}

__global__ void k_bnfin(const float* __restrict__ sums, const float* __restrict__ gamma,
                        const float* __restrict__ beta, float* __restrict__ scale,
                        float* __restrict__ shift, int N) {
  const int d = threadIdx.x;
  if (d >= CH) return;
  const float inv  = 1.0f / (float)N;
  const float mean = sums[d] * inv;
  const float var  = fmaf(-mean, mean, sums[CH + d] * inv);
  const float sc   = gamma[d] * rsqrtf(var + 1e-5f);
  scale[d] = sc;
  shift[d] = fmaf(-mean, sc, beta[d]);
}

extern "C" void kernel_launch(void* const* d_in, const int* in_sizes, int n_in,
                              void* d_out, int out_size, void* d_ws, size_t ws_size,
                              hipStream_t stream) {
  (void)n_in; (void)out_size; (void)ws_size;
  const float* x      = (const float*)d_in[0];
  const int*   eidx   = (const int*)d_in[1];
  const float* W1     = (const float*)d_in[2];
  const float* b1     = (const float*)d_in[3];
  const float* gamma1 = (const float*)d_in[4];
  const float* beta1  = (const float*)d_in[5];
  const float* W2     = (const float*)d_in[6];
  const float* b2     = (const float*)d_in[7];
  float* out = (float*)d_out;

  const int N = in_sizes[0] / CH;
  const int E = in_sizes[1] / 2;
  const int* rows = eidx;
  const int* cols = eidx + E;

  float* ws = (float*)d_ws;
  const size_t NH = (size_t)N * CH;
  float* hw   = ws;                            // [N*128] gemm result (reused)
  float* h1   = ws + NH;                       // [N*128] layer-1 features
  float* dis  = ws + 2 * NH;                   // [N]
  int*   degi = (int*)(ws + 2 * NH + (size_t)N);   // [N]
  float* sums = ws + 2 * NH + 2 * (size_t)N;   // [256] sum + sumsq
  float* scl  = sums + 256;                    // [128]
  float* sft  = scl + CH;                      // [128]

  const int nStrips    = (N + 15) / 16;
  const int gemmBlocks = (nStrips + 7) / 8;
  const int nBlocks    = (N + 255) / 256;
  const int eBlocks    = (E + 255) / 256;
  const int edgeBlocks = (E + 7) / 8;          // one wave (32 lanes) per edge
  const int seedBlocks = (N * 32 + 255) / 256;

  hipLaunchKernelGGL(k_init, dim3(nBlocks), dim3(256), 0, stream, degi, sums, N);
  hipLaunchKernelGGL(k_deg,  dim3(eBlocks), dim3(256), 0, stream, rows, degi, E);
  hipLaunchKernelGGL(k_dis,  dim3(nBlocks), dim3(256), 0, stream, degi, dis, N);

  // ---- Layer 1: GCNConv -> (BN stats) ----
  hipLaunchKernelGGL((k_gemm<false>), dim3(gemmBlocks), dim3(256), 0, stream,
                     x, W1, hw, (const float*)nullptr, (const float*)nullptr, N);
  hipLaunchKernelGGL(k_seed,  dim3(seedBlocks), dim3(256), 0, stream, hw, dis, b1, h1, N);
  hipLaunchKernelGGL(k_edges, dim3(edgeBlocks), dim3(256), 0, stream, rows, cols, dis, hw, h1, E);
  hipLaunchKernelGGL(k_bnstats, dim3(512), dim3(256), 0, stream, h1, sums, N);
  hipLaunchKernelGGL(k_bnfin, dim3(1), dim3(128), 0, stream, sums, gamma1, beta1, scl, sft, N);

  // ---- Layer 2: fused BN+ReLU in GEMM A-load -> GCNConv ----
  hipLaunchKernelGGL((k_gemm<true>), dim3(gemmBlocks), dim3(256), 0, stream,
                     h1, W2, hw, scl, sft, N);
  hipLaunchKernelGGL(k_seed,  dim3(seedBlocks), dim3(256), 0, stream, hw, dis, b2, out, N);
  hipLaunchKernelGGL(k_edges, dim3(edgeBlocks), dim3(256), 0, stream, rows, cols, dis, hw, out, E);
}